// GeometricAffine_56624848831039
// MI455X (gfx1250) — compile-verified
//
#include <hip/hip_runtime.h>
#include <math.h>

typedef __attribute__((ext_vector_type(2))) float v2f;
typedef __attribute__((ext_vector_type(8))) float v8f;

namespace {
constexpr int K_NN  = 24;
constexpr int CH    = 128;
constexpr int NPTS  = 4096;
constexpr int ROWS_PER_WAVE  = 32;
constexpr int WAVES          = 8;
constexpr int ROWS_PER_BLOCK = ROWS_PER_WAVE * WAVES;   // 256
constexpr int CHUNK   = 32;          // distance columns buffered per LDS pass
constexpr int LSTRIDE = CHUNK + 1;   // odd stride -> conflict-free row-per-lane b32 reads
constexpr int SCAN_BATCH = 8;        // LDS loads clustered per wait
constexpr float EPSV  = 1e-5f;
}

// Wave-private LDS ordering fence: DS ops from one wave execute in order in the
// LDS pipeline, so RAW through LDS within a wave only needs the compiler not to
// reorder; the explicit dscnt wait is conservative insurance.
__device__ __forceinline__ void wave_lds_fence() {
    asm volatile("s_wait_dscnt 0" ::: "memory");
}

// One kernel: WMMA distance tiles -> LDS -> per-lane register top-24 -> mean/std normalize.
__global__ __launch_bounds__(256) void
ga_topk_norm_kernel(const float* __restrict__ x,      // [B,C,N]
                    const float* __restrict__ xyz,    // [B,3,N]
                    const float* __restrict__ alpha,  // [C]
                    const float* __restrict__ beta,   // [C]
                    float* __restrict__ out)          // [B,C,N]
{
    __shared__ float lds[WAVES][ROWS_PER_WAVE][LSTRIDE];

    const int tid  = threadIdx.x;
    const int lane = tid & 31;
    const int wv   = tid >> 5;
    const int m    = lane & 15;
    const bool hi  = lane >= 16;

    const int rowBlocks = NPTS / ROWS_PER_BLOCK;            // 16
    const int b         = blockIdx.x / rowBlocks;
    const int blockRow  = (blockIdx.x % rowBlocks) * ROWS_PER_BLOCK;
    const int rBase     = blockRow + wv * ROWS_PER_WAVE;

    const float* xb = xyz + (size_t)b * 3 * NPTS;

    // ---- A fragments (16x4 f32 layout: lanes 0-15 K=0/1, lanes 16-31 K=2/3) ----
    // Pre-scaled by -2 so WMMA produces -2*dot; K=3 slot zero-padded.
    const int row0 = rBase + m;
    const int row1 = rBase + 16 + m;
    v2f a0, a1;
    if (!hi) {
        a0.x = -2.0f * xb[row0];            a0.y = -2.0f * xb[NPTS + row0];
        a1.x = -2.0f * xb[row1];            a1.y = -2.0f * xb[NPTS + row1];
    } else {
        a0.x = -2.0f * xb[2 * NPTS + row0]; a0.y = 0.0f;
        a1.x = -2.0f * xb[2 * NPTS + row1]; a1.y = 0.0f;
    }

    // ---- register-resident sorted top-K (smallest ranking key) ----
    float topd[K_NN];
    int   topi[K_NN];
#pragma unroll
    for (int k = 0; k < K_NN; ++k) { topd[k] = 3.4e38f; topi[k] = 0; }

    for (int chk = 0; chk < NPTS / CHUNK; ++chk) {
        const int colBase = chk * CHUNK;
#pragma unroll
        for (int t = 0; t < CHUNK / 16; ++t) {
            const int cb  = colBase + t * 16;
            const int col = cb + m;
            const float cx = xb[col];
            const float cy = xb[NPTS + col];
            const float cz = xb[2 * NPTS + col];
            v2f bf;
            if (!hi) { bf.x = cx; bf.y = cy; }
            else     { bf.x = cz; bf.y = 0.0f; }
            // ranking key = |dst|^2 - 2*dot  (row norm constant per row -> dropped)
            const float cn = cx * cx + cy * cy + cz * cz;

            v8f acc = {};
            v8f d0 = __builtin_amdgcn_wmma_f32_16x16x4_f32(
                         false, a0, false, bf, (short)0, acc, false, false);
            v8f d1 = __builtin_amdgcn_wmma_f32_16x16x4_f32(
                         false, a1, false, bf, (short)0, acc, false, false);

            // D layout: VGPR r = row r (lanes 0-15) / row r+8 (lanes 16-31), col = cb+(lane&15)
            const int lr = hi ? 8 : 0;
            const int cl = t * 16 + m;
#pragma unroll
            for (int r = 0; r < 8; ++r) {
                lds[wv][lr + r][cl]      = d0[r] + cn;
                lds[wv][lr + 16 + r][cl] = d1[r] + cn;
            }
        }
        wave_lds_fence();   // wave-private slab: no workgroup barrier needed

        // Each lane scans its own row. Batch LDS loads so one clause + one wait
        // covers SCAN_BATCH candidates instead of load->wait per element.
#pragma unroll
        for (int j = 0; j < CHUNK; j += SCAN_BATCH) {
            float v[SCAN_BATCH];
#pragma unroll
            for (int u = 0; u < SCAN_BATCH; ++u) v[u] = lds[wv][lane][j + u];
#pragma unroll
            for (int u = 0; u < SCAN_BATCH; ++u) {
                if (v[u] < topd[K_NN - 1]) {
                    topd[K_NN - 1] = v[u];
                    topi[K_NN - 1] = colBase + j + u;
#pragma unroll
                    for (int s = K_NN - 1; s > 0; --s) {
                        if (topd[s] < topd[s - 1]) {
                            float td = topd[s]; topd[s] = topd[s - 1]; topd[s - 1] = td;
                            int   ti = topi[s]; topi[s] = topi[s - 1]; topi[s - 1] = ti;
                        }
                    }
                }
            }
        }
        wave_lds_fence();   // reads done before next chunk's stores
    }

    // ---- phase 2: gather 24 neighbors per channel (L2-resident), mean/std, affine ----
    const int n = rBase + lane;
    const float* xrow = x   + (size_t)b * CH * NPTS;
    float*       orow = out + (size_t)b * CH * NPTS;

    for (int c = 0; c < CH; ++c) {
        const float* xc = xrow + (size_t)c * NPTS;
        float s = 0.0f, ss = 0.0f;
#pragma unroll
        for (int k = 0; k < K_NN; ++k) {
            const float g = xc[topi[k]];
            s += g;
            ss = fmaf(g, g, ss);
        }
        const float mean = s * (1.0f / K_NN);
        float var = (ss - (float)K_NN * mean * mean) * (1.0f / (K_NN - 1));
        var = fmaxf(var, 0.0f);
        const float stdv = sqrtf(var);
        const float xv = xc[n];
        orow[(size_t)c * NPTS + n] = (xv - mean) / (stdv + EPSV) * alpha[c] + beta[c];
    }
}

extern "C" void kernel_launch(void* const* d_in, const int* in_sizes, int n_in,
                              void* d_out, int out_size, void* d_ws, size_t ws_size,
                              hipStream_t stream) {
    (void)in_sizes; (void)n_in; (void)d_ws; (void)ws_size; (void)out_size;
    const float* x     = (const float*)d_in[0];   // [8,128,4096]
    const float* xyz   = (const float*)d_in[1];   // [8,3,4096]
    const float* alpha = (const float*)d_in[2];   // [128]
    const float* beta  = (const float*)d_in[3];   // [128]
    float* out = (float*)d_out;

    const int blocks = 8 * (NPTS / ROWS_PER_BLOCK);   // 8 batches * 16 row-blocks = 128
    ga_topk_norm_kernel<<<blocks, 256, 0, stream>>>(x, xyz, alpha, beta, out);
}